// OptNet_42365557408285
// MI455X (gfx1250) — compile-verified
//
#include <hip/hip_runtime.h>
#include <cstdint>

#define OPT_ITERS 25
#define OPT_SIGMA 0.1f
#define OPT_EPS   1e-8f

typedef float f32x2 __attribute__((ext_vector_type(2)));

// v_rcp_f32 (~1 ulp) + one Newton-Raphson step -> full fp32 accuracy,
// avoids the multi-instruction IEEE v_div_scale/v_div_fmas expansion.
__device__ __forceinline__ float fast_rcp(float x) {
  float r = __builtin_amdgcn_rcpf(x);
  r = r * __builtin_fmaf(-x, r, 2.0f);
  return r;
}

__global__ __launch_bounds__(256) void optnet_ipm_kernel(
    const float* __restrict__ Qm, const float* __restrict__ pv,
    const float* __restrict__ Gm, const float* __restrict__ hv,
    const float* __restrict__ Am, const float* __restrict__ bv,
    float* __restrict__ zout, int B)
{
  __shared__ float tile[256 * 3];   // [0..511] A pairs, [512..767] b
  const int tid = threadIdx.x;
  const int i   = blockIdx.x * 256 + tid;
  if (i >= B) return;

  float a0, a1, bb;
#if defined(__gfx1250__)
  {
    // Stage per-lane A (8B) and b (4B) into LDS via the CDNA5 async
    // global->LDS path (bypasses VGPRs, tracked by ASYNCcnt).
    const uint64_t gA = (uint64_t)(const void*)(Am + (size_t)i * 2);
    const uint64_t gB = (uint64_t)(const void*)(bv + i);
    // Flat LDS aperture: addr[31:0] is the wave-relative LDS byte offset.
    const unsigned lA = (unsigned)(uintptr_t)(void*)&tile[tid * 2];
    const unsigned lB = (unsigned)(uintptr_t)(void*)&tile[512 + tid];
    asm volatile(
        "global_load_async_to_lds_b64 %0, %2, off\n\t"
        "global_load_async_to_lds_b32 %1, %3, off\n\t"
        "s_wait_asynccnt 0x0"
        :
        : "v"(lA), "v"(lB), "v"(gA), "v"(gB)
        : "memory");
    a0 = tile[tid * 2 + 0];
    a1 = tile[tid * 2 + 1];
    bb = tile[512 + tid];
  }
#else
  a0 = Am[(size_t)i * 2 + 0];
  a1 = Am[(size_t)i * 2 + 1];
  bb = bv[i];
  (void)tile;
#endif

  // Shared problem constants (uniform address -> scalar loads).
  const float q00 = Qm[0], q01 = Qm[1], q10 = Qm[2], q11 = Qm[3];
  const float p0  = pv[0], p1  = pv[1];
  const float g00 = Gm[0], g01 = Gm[1], g10 = Gm[2], g11 = Gm[3];
  const float h0  = hv[0], h1  = hv[1];

  // IPM state: z (2), y (1), lam (2), s (2)
  float z0 = 0.f, z1 = 0.f, y = 0.f;
  float l0 = 1.f, l1 = 1.f, s0 = 1.f, s1 = 1.f;

#pragma unroll 1
  for (int it = 0; it < OPT_ITERS; ++it) {
    // Residuals
    float rz0 = z0 * q00 + z1 * q01 + p0 + a0 * y + l0 * g00 + l1 * g10;
    float rz1 = z0 * q10 + z1 * q11 + p1 + a1 * y + l0 * g01 + l1 * g11;
    float ry  = a0 * z0 + a1 * z1 - bb;
    float Gz0 = g00 * z0 + g01 * z1;
    float Gz1 = g10 * z0 + g11 * z1;
    float rs0 = Gz0 + s0 - h0;
    float rs1 = Gz1 + s1 - h1;
    float mu  = 0.5f * (l0 * s0 + l1 * s1);
    float rc0 = l0 * s0 - OPT_SIGMA * mu;
    float rc1 = l1 * s1 - OPT_SIGMA * mu;
    float r30 = -rc0 + l0 * rs0;          // rhs of row-3 block
    float r31 = -rc1 + l1 * rs1;

    // Condensed KKT: eliminate dlam (row-3 block diagonal in s+eps)
    float is0 = fast_rcp(s0 + OPT_EPS);
    float is1 = fast_rcp(s1 + OPT_EPS);
    float w0 = l0 * is0, w1 = l1 * is1;   // lam/(s+eps)
    float u0 = r30 * is0, u1 = r31 * is1;

    // H = Q + eps*I + G^T diag(w) G  (2x2 SPD)
    float H00 = q00 + OPT_EPS + g00 * g00 * w0 + g10 * g10 * w1;
    float H01 = q01           + g00 * g01 * w0 + g10 * g11 * w1;
    float H10 = q10           + g01 * g00 * w0 + g11 * g10 * w1;
    float H11 = q11 + OPT_EPS + g01 * g01 * w0 + g11 * g11 * w1;

    // rt = -rz - G^T u
    float rt0 = -rz0 - (g00 * u0 + g10 * u1);
    float rt1 = -rz1 - (g01 * u0 + g11 * u1);

    float idet = fast_rcp(H00 * H11 - H01 * H10);
    // k = H^-1 A^T,  m = H^-1 rt
    float k0 = ( H11 * a0 - H01 * a1) * idet;
    float k1 = (-H10 * a0 + H00 * a1) * idet;
    float m0 = ( H11 * rt0 - H01 * rt1) * idet;
    float m1 = (-H10 * rt0 + H00 * rt1) * idet;

    // Scalar Schur solve: (A H^-1 A^T - eps) dy = A H^-1 rt + ry
    float dy  = (a0 * m0 + a1 * m1 + ry) * fast_rcp(a0 * k0 + a1 * k1 - OPT_EPS);
    float dz0 = m0 - k0 * dy;
    float dz1 = m1 - k1 * dy;

    float Gdz0 = g00 * dz0 + g01 * dz1;
    float Gdz1 = g10 * dz0 + g11 * dz1;
    float dl0 = u0 + w0 * Gdz0;
    float dl1 = u1 + w1 * Gdz1;
    float ds0 = -rs0 - Gdz0;
    float ds1 = -rs1 - Gdz1;

    // Fraction-to-boundary step length
    const float BIG = 3.0e38f;
    float c0 = (dl0 < 0.f) ? (-l0 * fast_rcp(dl0)) : BIG;
    float c1 = (dl1 < 0.f) ? (-l1 * fast_rcp(dl1)) : BIG;
    float c2 = (ds0 < 0.f) ? (-s0 * fast_rcp(ds0)) : BIG;
    float c3 = (ds1 < 0.f) ? (-s1 * fast_rcp(ds1)) : BIG;
    float alpha = fminf(1.0f, 0.99f * fminf(fminf(c0, c1), fminf(c2, c3)));

    z0 += alpha * dz0; z1 += alpha * dz1; y += alpha * dy;
    l0 += alpha * dl0; l1 += alpha * dl1;
    s0 += alpha * ds0; s1 += alpha * ds1;
  }

  // Streamed output, no reuse -> non-temporal 8B store.
  f32x2 o; o.x = z0; o.y = z1;
  __builtin_nontemporal_store(o, (f32x2*)(zout + (size_t)i * 2));
}

extern "C" void kernel_launch(void* const* d_in, const int* in_sizes, int n_in,
                              void* d_out, int out_size, void* d_ws, size_t ws_size,
                              hipStream_t stream) {
  const float* Q = (const float*)d_in[0];
  const float* p = (const float*)d_in[1];
  const float* G = (const float*)d_in[2];
  const float* h = (const float*)d_in[3];
  const float* A = (const float*)d_in[4];
  const float* b = (const float*)d_in[5];
  float* out = (float*)d_out;

  const int B = in_sizes[5];            // b has B*NEQ = B elements
  const int block = 256;
  const int grid = (B + block - 1) / block;
  optnet_ipm_kernel<<<grid, block, 0, stream>>>(Q, p, G, h, A, b, out, B);
}